// NeuralBellmanFordNetwork_83674552860803
// MI455X (gfx1250) — compile-verified
//
#include <hip/hip_runtime.h>
#include <math.h>

#define NN 50000
#define NE 800000
#define D 32
#define KDIM 416          // 13*D
#define NUM_LAYERS 6
#define BOUNDARY 100.0f

typedef float v2f __attribute__((ext_vector_type(2)));
typedef float v8f __attribute__((ext_vector_type(8)));

// ---------------------------------------------------------------- WMMA helper
// CDNA5 fp32 matrix core: D(16x16,f32) = A(16x4,f32) x B(4x16,f32) + C
__device__ __forceinline__ v8f wmma_f32(v2f a, v2f b, v8f c) {
  return __builtin_amdgcn_wmma_f32_16x16x4_f32(
      /*neg_a=*/false, a, /*neg_b=*/false, b,
      /*c_mod=*/(short)0, c, /*reuse_a=*/false, /*reuse_b=*/false);
}

// ------------------------------------------------------- float atomic max/min
// sign-split trick: non-negative floats order like ints, negative like
// reversed unsigned -> lowers to global_atomic_max_i32 / min_u32 (no CAS loop)
__device__ __forceinline__ void atomicMaxF(float* a, float v) {
  if (v >= 0.0f) atomicMax((int*)a, __float_as_int(v));
  else           atomicMin((unsigned int*)a, (unsigned int)__float_as_int(v));
}
__device__ __forceinline__ void atomicMinF(float* a, float v) {
  if (v >= 0.0f) atomicMin((int*)a, __float_as_int(v));
  else           atomicMax((unsigned int*)a, (unsigned int)__float_as_int(v));
}

// ------------------------------------------------------------ pre-pass kernels
__global__ void k_pre_zero(float* degree, float* cnt, float* sumlog) {
  int n = blockIdx.x * blockDim.x + threadIdx.x;
  if (n < NN) { degree[n] = 0.0f; cnt[n] = 0.0f; }
  if (n == 0) sumlog[0] = 0.0f;
}

__global__ void k_edge_deg(const int* __restrict__ node_out,
                           const float* __restrict__ ew,
                           float* degree, float* cnt) {
  int e = blockIdx.x * blockDim.x + threadIdx.x;
  if (e >= NE) return;
  int dst = node_out[e];
  atomicAdd(&degree[dst], ew[e]);
  atomicAdd(&cnt[dst], 1.0f);
}

__global__ void k_node_deg(float* degree, float* cnt, float* scalev,
                           float* sumlog) {
  int n = blockIdx.x * blockDim.x + threadIdx.x;
  float lg = 0.0f;
  if (n < NN) {
    float dv = degree[n] + 1.0f;   // self loop weight 1
    degree[n] = dv;
    cnt[n] += 1.0f;                // self loop count
    lg = logf(dv);
    scalev[n] = lg;
  }
  // wave reduction to cut single-address atomic contention 32x
  #pragma unroll
  for (int off = 16; off > 0; off >>= 1) lg += __shfl_down(lg, off, 32);
  if ((threadIdx.x & 31) == 0) atomicAdd(sumlog, lg);
}

__global__ void k_scale_fin(float* scalev, const float* sumlog) {
  int n = blockIdx.x * blockDim.x + threadIdx.x;
  if (n >= NN) return;
  float mean = sumlog[0] / (float)NN;
  scalev[n] = scalev[n] / mean;
}

__global__ void k_init_li(float* li, const int* __restrict__ hptr) {
  int t = blockIdx.x * blockDim.x + threadIdx.x;
  if (t >= NN * D) return;
  li[t] = ((t >> 5) == hptr[0]) ? 0.0f : BOUNDARY;
}

// -------------------------------------------------------------- per layer prep
__global__ void k_rel(const float* __restrict__ q,
                      const float* __restrict__ rW,
                      const float* __restrict__ rb, float* rel) {
  int d = threadIdx.x;             // 32 threads
  float acc = rb[d];
  #pragma unroll
  for (int k = 0; k < D; ++k) acc = fmaf(q[k], rW[k * D + d], acc);
  rel[d] = acc;
}

__global__ void k_init_accum(float* s, float* ss, float* mx, float* mn) {
  int t = blockIdx.x * blockDim.x + threadIdx.x;
  if (t >= NN * D) return;
  s[t] = 0.0f; ss[t] = 0.0f;
  mx[t] = -__builtin_inff();
  mn[t] =  __builtin_inff();
}

// -------------------------------------------------------- edge scatter kernel
// one lane per (edge, feature); 32 lanes of a wave cover one edge's row
__global__ void k_edge_scatter(const int* __restrict__ node_in,
                               const int* __restrict__ node_out,
                               const float* __restrict__ ew,
                               const float* __restrict__ li,
                               const float* __restrict__ rel,
                               float* a_sum, float* a_sqs,
                               float* a_max, float* a_min) {
  long long t = (long long)blockIdx.x * blockDim.x + threadIdx.x;
  if (t >= (long long)NE * D) return;
  int e = (int)(t >> 5);
  int d = (int)(t & 31);
  int src = node_in[e];
  int dst = node_out[e];
  float w = ew[e];
  float m  = li[src * D + d] * rel[d];   // message
  float wm = m * w;                      // weighted message
  atomicAdd(&a_sum[dst * D + d], wm);
  atomicAdd(&a_sqs[dst * D + d], m * m * w);
  atomicMaxF(&a_max[dst * D + d], wm);
  atomicMinF(&a_min[dst * D + d], wm);
}

// ----------------------------------------------- predecessor (last layer only)
__device__ __forceinline__ float edge_score(const float* __restrict__ li,
                                            const float* __restrict__ rel,
                                            int src, float w) {
  float acc = 0.0f;
  #pragma unroll
  for (int d = 0; d < D; ++d) acc += li[src * D + d] * rel[d] * w;
  return acc;   // identical code in both passes -> bitwise-identical scores
}

__global__ void k_segmax_init(float* segmax, const int* __restrict__ hptr) {
  int n = blockIdx.x * blockDim.x + threadIdx.x;
  if (n >= NN) return;
  segmax[n] = (n == hptr[0]) ? 0.0f : (BOUNDARY * (float)D);  // self-loop score
}

__global__ void k_edge_score_max(const int* __restrict__ node_in,
                                 const int* __restrict__ node_out,
                                 const float* __restrict__ ew,
                                 const float* __restrict__ li,
                                 const float* __restrict__ rel,
                                 float* segmax) {
  int e = blockIdx.x * blockDim.x + threadIdx.x;
  if (e >= NE) return;
  atomicMaxF(&segmax[node_out[e]], edge_score(li, rel, node_in[e], ew[e]));
}

__global__ void k_pred_init(int* pred, const float* __restrict__ segmax,
                            const int* __restrict__ hptr) {
  int n = blockIdx.x * blockDim.x + threadIdx.x;
  if (n >= NN) return;
  float self = (n == hptr[0]) ? 0.0f : (BOUNDARY * (float)D);
  pred[n] = (self >= segmax[n]) ? n : NN;
}

__global__ void k_pred_edges(const int* __restrict__ node_in,
                             const int* __restrict__ node_out,
                             const float* __restrict__ ew,
                             const float* __restrict__ li,
                             const float* __restrict__ rel,
                             const float* __restrict__ segmax, int* pred) {
  int e = blockIdx.x * blockDim.x + threadIdx.x;
  if (e >= NE) return;
  int dst = node_out[e];
  if (edge_score(li, rel, node_in[e], ew[e]) >= segmax[dst])
    atomicMin(&pred[dst], node_in[e]);
}

// ---------------------------------------------------------- fused layer GEMM
// one wave per 16-row tile: build X[16,416] (layer_input ++ update) in LDS
// from the segment accumulators, then hidden = relu(X @ W + b) via
// v_wmma_f32_16x16x4_f32 (104 K-steps x 2 col tiles). N = 3125*16 exactly,
// so EXEC is all-ones around every WMMA (ISA requirement).
__global__ __launch_bounds__(32) void k_layer_gemm(
    const float* __restrict__ li,
    const float* __restrict__ a_sum, const float* __restrict__ a_sqs,
    const float* __restrict__ a_max, const float* __restrict__ a_min,
    const float* __restrict__ cnt, const float* __restrict__ scalev,
    const float* __restrict__ W /*416x32*/, const float* __restrict__ bias,
    const int* __restrict__ hptr, float* __restrict__ out /*N x 32*/) {
  __shared__ float Xs[16][KDIM + 1];   // stride 417 (odd mod 64) -> no conflicts
  const int lane = threadIdx.x;
  const int row0 = blockIdx.x * 16;
  const int h = hptr[0];

  // ---- build X tile: lane = feature d, loop over the 16 rows
  const int d = lane;
  for (int rr = 0; rr < 16; ++rr) {
    int n = row0 + rr;
    Xs[rr][d] = li[n * D + d];
    float b  = (n == h) ? 0.0f : BOUNDARY;     // self-loop (boundary) message
    float s  = a_sum[n * D + d] + b;
    float ss = a_sqs[n * D + d] + b * b;
    float mx = fmaxf(a_max[n * D + d], b);
    float mn = fminf(a_min[n * D + d], b);
    float c    = cnt[n];
    float mean = s / c;
    float sqm  = ss / c;
    float stdv = sqrtf(fmaxf(sqm - mean * mean, 1e-6f));
    float sc  = scalev[n];
    float inv = 1.0f / fmaxf(sc, 0.01f);
    float feats[4] = {mean, mx, mn, stdv};
    #pragma unroll
    for (int st = 0; st < 4; ++st) {
      float f = feats[st];
      int base = D + d * 12 + st * 3;   // k = 32 + (d*4+st)*3 + sidx
      Xs[rr][base + 0] = f;             // scale 1
      Xs[rr][base + 1] = f * sc;        // scale
      Xs[rr][base + 2] = f * inv;       // 1/clip(scale)
    }
  }
  __syncthreads();

  // ---- WMMA main loop
  const int hi = lane >> 4;   // 0/1 selects K pair {0,1} vs {2,3}
  const int ln = lane & 15;
  v8f c0 = {}; v8f c1 = {};
  #pragma unroll 4
  for (int kk = 0; kk < KDIM / 4; ++kk) {
    int k0 = kk * 4 + 2 * hi;
    v2f a; a.x = Xs[ln][k0]; a.y = Xs[ln][k0 + 1];
    v2f b0, b1;
    b0.x = W[k0 * D + ln];            b0.y = W[(k0 + 1) * D + ln];
    b1.x = W[k0 * D + 16 + ln];       b1.y = W[(k0 + 1) * D + 16 + ln];
    c0 = wmma_f32(a, b0, c0);
    c1 = wmma_f32(a, b1, c1);
  }

  // ---- epilogue: bias + relu, C layout: elem j -> row j+8*hi, col 16*ct+ln
  float bi0 = bias[ln], bi1 = bias[16 + ln];
  #pragma unroll
  for (int j = 0; j < 8; ++j) {
    int r = row0 + j + 8 * hi;
    out[r * D + ln]      = fmaxf(c0[j] + bi0, 0.0f);
    out[r * D + 16 + ln] = fmaxf(c1[j] + bi1, 0.0f);
  }
}

// -------------------------------------------------------------- final MLP head
// score = relu([hidden, query] @ W1 + b1) @ W2 + b2, fused: 16 K-steps x 4 col
// tiles of f32 WMMA, then the W2 dot + cross-lane reduction in the epilogue.
__global__ __launch_bounds__(32) void k_mlp(
    const float* __restrict__ li, const float* __restrict__ query,
    const float* __restrict__ W1 /*64x64*/, const float* __restrict__ b1,
    const float* __restrict__ W2 /*64*/, const float* __restrict__ b2,
    float* __restrict__ score) {
  __shared__ float Xs[16][65];        // stride 65 -> conflict-free
  const int lane = threadIdx.x;
  const int row0 = blockIdx.x * 16;
  for (int rr = 0; rr < 16; ++rr) {
    Xs[rr][lane]     = li[(row0 + rr) * D + lane];
    Xs[rr][D + lane] = query[lane];
  }
  __syncthreads();

  const int hi = lane >> 4;
  const int ln = lane & 15;
  v8f c[4] = {{}, {}, {}, {}};
  #pragma unroll
  for (int kk = 0; kk < 16; ++kk) {
    int k0 = kk * 4 + 2 * hi;
    v2f a; a.x = Xs[ln][k0]; a.y = Xs[ln][k0 + 1];
    #pragma unroll
    for (int ct = 0; ct < 4; ++ct) {
      v2f b; b.x = W1[k0 * 64 + ct * 16 + ln];
             b.y = W1[(k0 + 1) * 64 + ct * 16 + ln];
      c[ct] = wmma_f32(a, b, c[ct]);
    }
  }

  float p[8];
  #pragma unroll
  for (int j = 0; j < 8; ++j) p[j] = 0.0f;
  #pragma unroll
  for (int ct = 0; ct < 4; ++ct) {
    int col = ct * 16 + ln;
    float bb = b1[col], w2 = W2[col];
    #pragma unroll
    for (int j = 0; j < 8; ++j)
      p[j] += fmaxf(c[ct][j] + bb, 0.0f) * w2;
  }
  #pragma unroll
  for (int j = 0; j < 8; ++j) {
    p[j] += __shfl_xor(p[j], 1, 16);
    p[j] += __shfl_xor(p[j], 2, 16);
    p[j] += __shfl_xor(p[j], 4, 16);
    p[j] += __shfl_xor(p[j], 8, 16);
  }
  if (ln == 0) {
    float bb2 = b2[0];
    #pragma unroll
    for (int j = 0; j < 8; ++j)
      score[row0 + 8 * hi + j] = p[j] + bb2;
  }
}

__global__ void k_write_pred(const int* __restrict__ pred, int* outp) {
  int n = blockIdx.x * blockDim.x + threadIdx.x;
  if (n < NN) outp[NN + n] = pred[n];
}

// ------------------------------------------------------------------- launcher
extern "C" void kernel_launch(void* const* d_in, const int* in_sizes, int n_in,
                              void* d_out, int out_size, void* d_ws,
                              size_t ws_size, hipStream_t stream) {
  const int*   node_in  = (const int*)d_in[0];
  const int*   node_out = (const int*)d_in[1];
  const float* edge_w   = (const float*)d_in[2];
  const int*   h_ptr    = (const int*)d_in[3];
  const float* query    = (const float*)d_in[4];
  const float* rel_W    = (const float*)d_in[5];
  const float* rel_b    = (const float*)d_in[6];
  const float* lin_W    = (const float*)d_in[7];
  const float* lin_b    = (const float*)d_in[8];
  const float* mlp_W1   = (const float*)d_in[9];
  const float* mlp_b1   = (const float*)d_in[10];
  const float* mlp_W2   = (const float*)d_in[11];
  const float* mlp_b2   = (const float*)d_in[12];

  float* out = (float*)d_out;
  float* ws  = (float*)d_ws;
  const size_t N = NN;
  float* degree = ws;                 // N
  float* cnt    = ws + N;             // N
  float* scalev = ws + 2 * N;         // N
  float* sumlog = ws + 3 * N;         // 1 (padded)
  float* rel    = ws + 3 * N + 16;    // 32 (padded to 64)
  float* segmax = ws + 3 * N + 64;    // N
  int*   pred   = (int*)(ws + 4 * N + 64);  // N
  float* li_a   = ws + 5 * N + 64;    // N*32
  float* li_b   = li_a + 32 * N;      // N*32
  float* a_sum  = li_b + 32 * N;      // N*32
  float* a_sqs  = a_sum + 32 * N;     // N*32
  float* a_max  = a_sqs + 32 * N;     // N*32
  float* a_min  = a_max + 32 * N;     // N*32  -> total ~39.4 MB

  const int TB = 256;
  const int gN   = (NN + TB - 1) / TB;
  const int gE   = (NE + TB - 1) / TB;
  const int gND  = (NN * D + TB - 1) / TB;
  const int gED  = (int)(((long long)NE * D + TB - 1) / TB);
  const int gTile = NN / 16;   // 3125, exact

  // ---- preamble: degrees, counts, scale, boundary layer input
  k_pre_zero<<<gN, TB, 0, stream>>>(degree, cnt, sumlog);
  k_edge_deg<<<gE, TB, 0, stream>>>(node_out, edge_w, degree, cnt);
  k_node_deg<<<gN, TB, 0, stream>>>(degree, cnt, scalev, sumlog);
  k_scale_fin<<<gN, TB, 0, stream>>>(scalev, sumlog);
  k_init_li<<<gND, TB, 0, stream>>>(li_a, h_ptr);

  float* cur = li_a;
  float* nxt = li_b;
  for (int i = 0; i < NUM_LAYERS; ++i) {
    k_rel<<<1, 32, 0, stream>>>(query, rel_W + (size_t)i * D * D,
                                rel_b + (size_t)i * D, rel);
    k_init_accum<<<gND, TB, 0, stream>>>(a_sum, a_sqs, a_max, a_min);
    k_edge_scatter<<<gED, TB, 0, stream>>>(node_in, node_out, edge_w, cur, rel,
                                           a_sum, a_sqs, a_max, a_min);
    if (i == NUM_LAYERS - 1) {
      k_segmax_init<<<gN, TB, 0, stream>>>(segmax, h_ptr);
      k_edge_score_max<<<gE, TB, 0, stream>>>(node_in, node_out, edge_w, cur,
                                              rel, segmax);
      k_pred_init<<<gN, TB, 0, stream>>>(pred, segmax, h_ptr);
      k_pred_edges<<<gE, TB, 0, stream>>>(node_in, node_out, edge_w, cur, rel,
                                          segmax, pred);
    }
    k_layer_gemm<<<gTile, 32, 0, stream>>>(
        cur, a_sum, a_sqs, a_max, a_min, cnt, scalev,
        lin_W + (size_t)i * KDIM * D, lin_b + (size_t)i * D, h_ptr, nxt);
    float* t = cur; cur = nxt; nxt = t;
  }

  k_mlp<<<gTile, 32, 0, stream>>>(cur, query, mlp_W1, mlp_b1, mlp_W2, mlp_b2,
                                  out);
  k_write_pred<<<gN, TB, 0, stream>>>(pred, (int*)d_out);
}